// LANTModel_33741263078159
// MI455X (gfx1250) — compile-verified
//
#include <hip/hip_runtime.h>
#include <math.h>

#define N_NODES 100000
#define N_EDGES 1600000
#define F_IN    64
#define HEADS   2
#define D_HEAD  64
#define F_OUT   128   // HEADS * D_HEAD

typedef __attribute__((ext_vector_type(2))) float v2f;
typedef __attribute__((ext_vector_type(8))) float v8f;

// ---------------------------------------------------------------------------
// float atomic max via sign-magnitude trick (init must be -inf)
__device__ __forceinline__ void atomic_max_float(float* addr, float v) {
  if (v >= 0.0f) {
    atomicMax(reinterpret_cast<int*>(addr), __float_as_int(v));
  } else {
    atomicMin(reinterpret_cast<unsigned int*>(addr), __float_as_uint(v));
  }
}

// ---------------------------------------------------------------------------
// w_s[h*64+k] = sum_d W_src[k, h*64+d] * a_src[h,d]   (and same for dst side)
// also zeroes the 128-float summary accumulator.
__global__ void dgi_prep_kernel(const float* __restrict__ W_src,
                                const float* __restrict__ W_dst,
                                const float* __restrict__ a_src,
                                const float* __restrict__ a_dst,
                                float* __restrict__ wsv,
                                float* __restrict__ wdv,
                                float* __restrict__ sumbuf) {
  int t = threadIdx.x;            // 0..127  -> (h,k)
  int h = t >> 6;
  int k = t & 63;
  float acc_s = 0.f, acc_d = 0.f;
  #pragma unroll 8
  for (int d = 0; d < D_HEAD; ++d) {
    acc_s += W_src[k * F_OUT + h * D_HEAD + d] * a_src[h * D_HEAD + d];
    acc_d += W_dst[k * F_OUT + h * D_HEAD + d] * a_dst[h * D_HEAD + d];
  }
  wsv[t] = acc_s;
  wdv[t] = acc_d;
  sumbuf[t] = 0.f;
}

// ---------------------------------------------------------------------------
// al_s[n,h] = x[n] . wsv[h]; al_d[n,h] = x[n] . wdv[h]. One wave32 per node.
__global__ void dgi_al_kernel(const float* __restrict__ x,
                              const float* __restrict__ wsv,
                              const float* __restrict__ wdv,
                              float* __restrict__ al_s,
                              float* __restrict__ al_d) {
  int wave = threadIdx.x >> 5;
  int lane = threadIdx.x & 31;
  int n = blockIdx.x * 8 + wave;        // 12500 blocks * 8 waves == 100000
  const float* xr = x + (size_t)n * F_IN;
  float x0 = xr[lane];
  float x1 = xr[lane + 32];
  float s0 = x0 * wsv[lane]      + x1 * wsv[lane + 32];
  float s1 = x0 * wsv[64 + lane] + x1 * wsv[96 + lane];
  float d0 = x0 * wdv[lane]      + x1 * wdv[lane + 32];
  float d1 = x0 * wdv[64 + lane] + x1 * wdv[96 + lane];
  #pragma unroll
  for (int off = 16; off > 0; off >>= 1) {
    s0 += __shfl_xor(s0, off, 32);
    s1 += __shfl_xor(s1, off, 32);
    d0 += __shfl_xor(d0, off, 32);
    d1 += __shfl_xor(d1, off, 32);
  }
  if (lane == 0) {
    al_s[n * 2 + 0] = s0;
    al_s[n * 2 + 1] = s1;
    al_d[n * 2 + 0] = d0;
    al_d[n * 2 + 1] = d1;
  }
}

// ---------------------------------------------------------------------------
// xs = x @ W_src : [100000,64] x [64,128], fp32 WMMA 16x16x4.
// 8 waves / block, wave w owns the 16x16 tile at cols w*16, rows blockIdx*16.
__global__ void dgi_gemm_kernel(const float* __restrict__ x,
                                const float* __restrict__ W,   // [64,128] row-major
                                float* __restrict__ xs) {
  const int wave = threadIdx.x >> 5;       // 0..7 -> column strip
  const int lane = threadIdx.x & 31;
  const int row0 = blockIdx.x * 16;        // 6250 blocks * 16 == 100000
  const int col0 = wave * 16;
  const int m  = lane & 15;                // A-row / B-col within tile
  const int hi = lane >> 4;                // lane half selects K pair
  v8f c = {0.f, 0.f, 0.f, 0.f, 0.f, 0.f, 0.f, 0.f};
  const float* xrow = x + (size_t)(row0 + m) * F_IN;
  #pragma unroll
  for (int k0 = 0; k0 < F_IN; k0 += 4) {
    const int kb = k0 + hi * 2;
    v2f a;                                  // A 16x4: lanes 0-15 K={k0,k0+1}, 16-31 K={k0+2,k0+3}
    a.x = xrow[kb];
    a.y = xrow[kb + 1];
    v2f b;                                  // B 4x16: mirrored K striping, N across lanes
    b.x = W[(size_t)kb * F_OUT + col0 + m];
    b.y = W[(size_t)(kb + 1) * F_OUT + col0 + m];
    c = __builtin_amdgcn_wmma_f32_16x16x4_f32(false, a, false, b, (short)0, c,
                                              false, false);
  }
  // C/D layout: VGPR v -> row v (lanes 0-15) / row v+8 (lanes 16-31), col = lane&15
  #pragma unroll
  for (int v = 0; v < 8; ++v) {
    int row = row0 + v + hi * 8;
    xs[(size_t)row * F_OUT + col0 + m] = c[v];
  }
}

// ---------------------------------------------------------------------------
__global__ void dgi_init_md_kernel(float* __restrict__ mbuf,
                                   float* __restrict__ den) {
  int i = blockIdx.x * blockDim.x + threadIdx.x;
  if (i < N_NODES * HEADS) {
    mbuf[i] = __uint_as_float(0xFF800000u);   // -inf
    den[i]  = 0.f;
  }
}

// ---------------------------------------------------------------------------
// e[e,h] = leaky_relu(al_s[P(src)] + al_d[P(dst)]); segment max into m[dst,h].
__global__ void dgi_logits_kernel(const int* __restrict__ src,
                                  const int* __restrict__ dst,
                                  const int* __restrict__ perm,
                                  const float* __restrict__ al_s,
                                  const float* __restrict__ al_d,
                                  float* __restrict__ e_buf,
                                  float* __restrict__ mbuf,
                                  int neg) {
  int i = blockIdx.x * blockDim.x + threadIdx.x;   // (edge, head)
  if (i >= N_EDGES * HEADS) return;
  int e = i >> 1;
  int h = i & 1;
  int s = src[e];
  int d = dst[e];
  int sp = neg ? perm[s] : s;
  int dp = neg ? perm[d] : d;
  float v = al_s[sp * 2 + h] + al_d[dp * 2 + h];
  v = (v >= 0.f) ? v : 0.2f * v;                   // leaky_relu(0.2)
  e_buf[i] = v;
  atomic_max_float(&mbuf[d * 2 + h], v);
}

// ---------------------------------------------------------------------------
// ex = exp(e - m[dst]); segment sum into den[dst,h]; e_buf <- ex.
__global__ void dgi_exp_kernel(const int* __restrict__ dst,
                               float* __restrict__ e_buf,
                               const float* __restrict__ mbuf,
                               float* __restrict__ den) {
  int i = blockIdx.x * blockDim.x + threadIdx.x;
  if (i >= N_EDGES * HEADS) return;
  int e = i >> 1;
  int h = i & 1;
  int d = dst[e];
  float ex = expf(e_buf[i] - mbuf[d * 2 + h]);
  e_buf[i] = ex;
  atomicAdd(&den[d * 2 + h], ex);
}

// ---------------------------------------------------------------------------
// out[dst] += alpha * xs[P(src)]. 128 threads per edge, 2 edges per block.
__global__ void dgi_scatter_kernel(const int* __restrict__ src,
                                   const int* __restrict__ dst,
                                   const int* __restrict__ perm,
                                   const float* __restrict__ e_buf,  // ex
                                   const float* __restrict__ den,
                                   const float* __restrict__ xs,
                                   float* __restrict__ out_z,
                                   int neg) {
  int e = blockIdx.x * 2 + (threadIdx.x >> 7);     // 800000 blocks * 2 == E
  int f = threadIdx.x & 127;
  int h = f >> 6;
  int s = src[e];
  int d = dst[e];
  int sp = neg ? perm[s] : s;
  float alpha = e_buf[e * 2 + h] / (den[d * 2 + h] + 1e-16f);
  atomicAdd(&out_z[(size_t)d * F_OUT + f], xs[(size_t)sp * F_OUT + f] * alpha);
}

// ---------------------------------------------------------------------------
// z = prelu(z + bias); optionally accumulate column sums for the summary.
// stride is a multiple of 128 so each thread keeps a fixed feature f.
__global__ void dgi_finalize_kernel(float* __restrict__ z,
                                    const float* __restrict__ bias,
                                    const float* __restrict__ pw,
                                    float* __restrict__ sumbuf,
                                    int do_sum) {
  int tid = blockIdx.x * blockDim.x + threadIdx.x;
  int stride = gridDim.x * blockDim.x;             // multiple of 128
  int f = tid & 127;
  float bv = bias[f];
  float p  = pw[f];
  float local = 0.f;
  const int total = N_NODES * F_OUT;
  for (int i = tid; i < total; i += stride) {
    float v = z[i] + bv;
    v = (v >= 0.f) ? v : p * v;
    z[i] = v;
    local += v;
  }
  if (do_sum) atomicAdd(&sumbuf[f], local);
}

// ---------------------------------------------------------------------------
__global__ void dgi_summary_kernel(const float* __restrict__ sumbuf,
                                   float* __restrict__ out) {
  int f = threadIdx.x;   // 128 threads
  float s = sumbuf[f] * (1.0f / (float)N_NODES);
  out[f] = 1.0f / (1.0f + expf(-s));
}

// ---------------------------------------------------------------------------
extern "C" void kernel_launch(void* const* d_in, const int* in_sizes, int n_in,
                              void* d_out, int out_size, void* d_ws, size_t ws_size,
                              hipStream_t stream) {
  const float* x      = (const float*)d_in[0];
  const int*   ei     = (const int*)d_in[1];       // [2, E]
  const int*   perm   = (const int*)d_in[2];
  const float* W_src  = (const float*)d_in[3];     // [64,128]
  const float* W_dst  = (const float*)d_in[4];
  const float* a_src  = (const float*)d_in[5];     // [2,64]
  const float* a_dst  = (const float*)d_in[6];
  const float* bias   = (const float*)d_in[7];
  const float* preluw = (const float*)d_in[8];
  const int* src = ei;
  const int* dst = ei + N_EDGES;

  // workspace layout (floats): ~67 MB total
  float* ws     = (float*)d_ws;
  float* xs     = ws;                               // N*128
  float* al_s   = xs   + (size_t)N_NODES * F_OUT;   // N*2
  float* al_d   = al_s + (size_t)N_NODES * HEADS;   // N*2
  float* e_buf  = al_d + (size_t)N_NODES * HEADS;   // E*2
  float* mbuf   = e_buf + (size_t)N_EDGES * HEADS;  // N*2
  float* den    = mbuf + (size_t)N_NODES * HEADS;   // N*2
  float* wsv    = den  + (size_t)N_NODES * HEADS;   // 128
  float* wdv    = wsv + F_OUT;                      // 128
  float* sumbuf = wdv + F_OUT;                      // 128

  float* out    = (float*)d_out;
  float* pos_z  = out;
  float* neg_z  = out + (size_t)N_NODES * F_OUT;
  float* summ   = out + (size_t)2 * N_NODES * F_OUT;

  // d_out doubles as the segment-sum accumulator -> zero it.
  hipMemsetAsync(d_out, 0, (size_t)out_size * sizeof(float), stream);

  dgi_prep_kernel<<<1, 128, 0, stream>>>(W_src, W_dst, a_src, a_dst, wsv, wdv, sumbuf);
  dgi_al_kernel<<<N_NODES / 8, 256, 0, stream>>>(x, wsv, wdv, al_s, al_d);
  dgi_gemm_kernel<<<N_NODES / 16, 256, 0, stream>>>(x, W_src, xs);

  const int eh_blocks = (N_EDGES * HEADS + 255) / 256;
  const int md_blocks = (N_NODES * HEADS + 255) / 256;

  for (int pass = 0; pass < 2; ++pass) {
    float* z = (pass == 0) ? pos_z : neg_z;
    dgi_init_md_kernel<<<md_blocks, 256, 0, stream>>>(mbuf, den);
    dgi_logits_kernel<<<eh_blocks, 256, 0, stream>>>(src, dst, perm, al_s, al_d,
                                                     e_buf, mbuf, pass);
    dgi_exp_kernel<<<eh_blocks, 256, 0, stream>>>(dst, e_buf, mbuf, den);
    dgi_scatter_kernel<<<N_EDGES / 2, 256, 0, stream>>>(src, dst, perm, e_buf, den,
                                                        xs, z, pass);
    dgi_finalize_kernel<<<1024, 256, 0, stream>>>(z, bias, preluw, sumbuf,
                                                  pass == 0 ? 1 : 0);
  }
  dgi_summary_kernel<<<1, 128, 0, stream>>>(sumbuf, summ);
}